// MultiScalePUMP_43954695307635
// MI455X (gfx1250) — compile-verified
//
#include <hip/hip_runtime.h>
#include <math.h>

typedef __attribute__((ext_vector_type(2))) float v2f;
typedef __attribute__((ext_vector_type(4))) float v4f;
typedef __attribute__((ext_vector_type(8))) float v8f;

#define N_CORRES 4096
#define GRID_HW 256
#define G_TOTAL (GRID_HW * GRID_HW)
#define WAVES_PER_BLOCK 8
#define TILE_N 16                      // grid points per wave (WMMA N)
#define TILE_M 16                      // pos rows per WMMA tile (WMMA M)
#define N_TILES (N_CORRES / TILE_M)    // 256 WMMA iterations per wave
#define BLOCK_THREADS 256

// lane L <-> lane L^16 (group-of-32 swizzle: and=0x1f, or=0, xor=0x10)
#define SWZ_SWAPX16 0x401F

__device__ __forceinline__ void topk_insert(float dv, int di,
                                            float& v0, float& v1, float& v2, float& v3,
                                            int& i0, int& i1, int& i2, int& i3) {
    // replace last element if smaller, then one bubble pass restores sort
    bool ins = dv < v3;
    v3 = ins ? dv : v3;  i3 = ins ? di : i3;
    bool s2 = v3 < v2;  float tv; int ti;
    tv = s2 ? v3 : v2;  v3 = s2 ? v2 : v3;  v2 = tv;
    ti = s2 ? i3 : i2;  i3 = s2 ? i2 : i3;  i2 = ti;
    bool s1 = v2 < v1;
    tv = s1 ? v2 : v1;  v2 = s1 ? v1 : v2;  v1 = tv;
    ti = s1 ? i2 : i1;  i2 = s1 ? i1 : i2;  i1 = ti;
    bool s0 = v1 < v0;
    tv = s0 ? v1 : v0;  v1 = s0 ? v0 : v1;  v0 = tv;
    ti = s0 ? i1 : i0;  i1 = s0 ? i0 : i1;  i0 = ti;
}

__global__ __launch_bounds__(BLOCK_THREADS)
void MultiScalePUMP_kernel(const float* __restrict__ pos,     // (4096,4)
                           const float* __restrict__ scores,  // (4096,)
                           const float* __restrict__ grid,    // (65536,2)
                           const float* __restrict__ accu,    // (65536,6)
                           const float* __restrict__ scale_p, // (1,)
                           const float* __restrict__ code_p,  // (1,)
                           float* __restrict__ out)           // (65536,6)
{
    __shared__ v4f s_pos[N_CORRES];   // {-2x, -2y, |p|^2, 0} per correspondence

    const int tid = threadIdx.x;

    // ---- stage pos into LDS (coalesced), precompute WMMA A-side terms ----
    for (int k = 0; k < N_CORRES / BLOCK_THREADS; ++k) {
        int i = tid + k * BLOCK_THREADS;
        v4f p = ((const v4f*)pos)[i];
        v4f e;
        e.x = -2.0f * p.x;
        e.y = -2.0f * p.y;
        e.z = p.x * p.x + p.y * p.y;
        e.w = 0.0f;
        s_pos[i] = e;
    }
    __syncthreads();

    const int wave  = tid >> 5;
    const int lane  = tid & 31;
    const int lhalf = lane >> 4;        // 0: lanes 0-15, 1: lanes 16-31
    const int l16   = lane & 15;
    const int gbase = (blockIdx.x * WAVES_PER_BLOCK + wave) * TILE_N;
    const int g     = gbase + l16;      // this lane's grid point (WMMA N)

    // ---- B matrix (constant over the whole loop) ----
    // cols of B are grid points: [gx, gy, 1, 0]^T.
    // lanes 0-15 hold K=0,1 rows -> (gx,gy); lanes 16-31 hold K=2,3 -> (1,0).
    float gx = grid[2 * g + 0];
    float gy = grid[2 * g + 1];
    float gn2 = gx * gx + gy * gy;
    v2f b;
    b.x = lhalf ? 1.0f : gx;
    b.y = lhalf ? 0.0f : gy;

    // running top-4 smallest d^2 (ascending), with pos indices
    float v0 = INFINITY, v1 = INFINITY, v2 = INFINITY, v3 = INFINITY;
    int   i0 = 0, i1 = 0, i2 = 0, i3 = 0;

    const float* sp = (const float*)s_pos;
    const int aoff = lhalf ? 2 : 0;

    // software pipeline: prefetch A for tile t+1 before consuming tile t
    v2f a_cur = *(const v2f*)(sp + l16 * 4 + aoff);

#pragma unroll 2
    for (int t = 0; t < N_TILES; ++t) {
        int tn = (t + 1) & (N_TILES - 1);   // last trip redundantly reloads tile 0
        v2f a_nxt = *(const v2f*)(sp + (tn * TILE_M + l16) * 4 + aoff);

        // C = |g_n|^2 broadcast (per-lane constant): D = |p|^2 + |g|^2 - 2 p.g
        v8f c = { gn2, gn2, gn2, gn2, gn2, gn2, gn2, gn2 };
        v8f d = __builtin_amdgcn_wmma_f32_16x16x4_f32(
            false, a_cur, false, b, (short)0, c, false, false);

        // cheap reject: if the best of this tile can't enter the top-4, skip.
        // Same strict '<' as the insert, so the skip is exactly equivalent.
        float m = fminf(fminf(fminf(d[0], d[1]), fminf(d[2], d[3])),
                        fminf(fminf(d[4], d[5]), fminf(d[6], d[7])));
        if (m < v3) {
            // lane holds d^2 for pos rows M = lhalf*8 + j, fixed grid point N
            int base = t * TILE_M + lhalf * 8;
#pragma unroll
            for (int j = 0; j < 8; ++j) {
                topk_insert(d[j], base + j, v0, v1, v2, v3, i0, i1, i2, i3);
            }
        }
        a_cur = a_nxt;
    }

    // ---- merge the two half-lane candidate streams for each grid point ----
    float pv0 = __int_as_float(__builtin_amdgcn_ds_swizzle(__float_as_int(v0), SWZ_SWAPX16));
    float pv1 = __int_as_float(__builtin_amdgcn_ds_swizzle(__float_as_int(v1), SWZ_SWAPX16));
    float pv2 = __int_as_float(__builtin_amdgcn_ds_swizzle(__float_as_int(v2), SWZ_SWAPX16));
    float pv3 = __int_as_float(__builtin_amdgcn_ds_swizzle(__float_as_int(v3), SWZ_SWAPX16));
    int   pj0 = __builtin_amdgcn_ds_swizzle(i0, SWZ_SWAPX16);
    int   pj1 = __builtin_amdgcn_ds_swizzle(i1, SWZ_SWAPX16);
    int   pj2 = __builtin_amdgcn_ds_swizzle(i2, SWZ_SWAPX16);
    int   pj3 = __builtin_amdgcn_ds_swizzle(i3, SWZ_SWAPX16);
    topk_insert(pv0, pj0, v0, v1, v2, v3, i0, i1, i2, i3);
    topk_insert(pv1, pj1, v0, v1, v2, v3, i0, i1, i2, i3);
    topk_insert(pv2, pj2, v0, v1, v2, v3, i0, i1, i2, i3);
    topk_insert(pv3, pj3, v0, v1, v2, v3, i0, i1, i2, i3);

    // ---- final scoring + conditional accumulator update (lanes 0-15) ----
    if (lane < 16) {
        float scale = scale_p[0];
        float code  = code_p[0];
        float dist_max = 8.0f * scale - 1e-7f;

        float d0 = sqrtf(fmaxf(v0, 1e-12f));
        float d1 = sqrtf(fmaxf(v1, 1e-12f));
        float d2 = sqrtf(fmaxf(v2, 1e-12f));
        float d3 = sqrtf(fmaxf(v3, 1e-12f));
        float thr = 2.0f * d0;

        float c0 = (d0 <= thr && d0 < dist_max) ? scores[i0] : -INFINITY;
        float c1 = (d1 <= thr && d1 < dist_max) ? scores[i1] : -INFINITY;
        float c2 = (d2 <= thr && d2 < dist_max) ? scores[i2] : -INFINITY;
        float c3 = (d3 <= thr && d3 < dist_max) ? scores[i3] : -INFINITY;

        // argmax keeps first occurrence (candidates are distance-ordered)
        float best = c0; int sel = i0;
        if (c1 > best) { best = c1; sel = i1; }
        if (c2 > best) { best = c2; sel = i2; }
        if (c3 > best) { best = c3; sel = i3; }

        const float* arow = accu + (size_t)g * 6;
        float a4 = arow[4];
        float o0, o1, o2, o3, o4, o5;
        if (best > a4) {
            const float* pr = pos + (size_t)sel * 4;
            o0 = pr[0]; o1 = pr[1]; o2 = pr[2]; o3 = pr[3];
            o4 = best;  o5 = code;
        } else {
            o0 = arow[0]; o1 = arow[1]; o2 = arow[2]; o3 = arow[3];
            o4 = a4;      o5 = arow[5];
        }
        float* orow = out + (size_t)g * 6;
        orow[0] = o0; orow[1] = o1; orow[2] = o2;
        orow[3] = o3; orow[4] = o4; orow[5] = o5;
    }
}

extern "C" void kernel_launch(void* const* d_in, const int* in_sizes, int n_in,
                              void* d_out, int out_size, void* d_ws, size_t ws_size,
                              hipStream_t stream) {
    (void)in_sizes; (void)n_in; (void)out_size; (void)d_ws; (void)ws_size;
    const float* pos    = (const float*)d_in[0];
    const float* scores = (const float*)d_in[1];
    const float* grid   = (const float*)d_in[2];
    const float* accu   = (const float*)d_in[3];
    const float* scale  = (const float*)d_in[4];
    const float* code   = (const float*)d_in[5];
    float* out = (float*)d_out;

    dim3 gridDim(G_TOTAL / (WAVES_PER_BLOCK * TILE_N)); // 512 blocks
    dim3 blockDim(BLOCK_THREADS);                       // 8 waves of 32
    MultiScalePUMP_kernel<<<gridDim, blockDim, 0, stream>>>(
        pos, scores, grid, accu, scale, code, out);
}